// MeshConvPoint_35132832481370
// MI455X (gfx1250) — compile-verified
//
#include <hip/hip_runtime.h>
#include <hip/hip_bf16.h>

typedef __attribute__((ext_vector_type(2))) float v2f;
typedef __attribute__((ext_vector_type(4))) float v4f;
typedef __attribute__((ext_vector_type(8))) float v8f;

#define C_IN   128
#define C_OUT  256
#define KNBR   7
#define KD     (C_IN * KNBR)   // 896
#define TILE_N 64              // vertices per workgroup
#define FSTR   132             // LDS row stride in dwords (128 + 4 pad -> conflict-free b64)

// ---------------------------------------------------------------------------
// Kernel 1: transpose x [B,C,N] -> xT [B,N,C] so gathers are 512B-contiguous.
// x is read exactly once -> nontemporal loads keep it out of L2's way.
// ---------------------------------------------------------------------------
__global__ __launch_bounds__(256)
void transpose_x_kernel(const float* __restrict__ x, float* __restrict__ xT, int Nv) {
  __shared__ float tile[32][33];
  const int b  = blockIdx.z;
  const int c0 = blockIdx.y * 32;
  const int n0 = blockIdx.x * 32;
  const int tx = threadIdx.x, ty = threadIdx.y;
  const float* xb  = x  + (size_t)b * C_IN * Nv;
  float*       xTb = xT + (size_t)b * Nv * C_IN;
#pragma unroll
  for (int i = 0; i < 32; i += 8)
    tile[ty + i][tx] = __builtin_nontemporal_load(&xb[(size_t)(c0 + ty + i) * Nv + (n0 + tx)]);
  __syncthreads();
#pragma unroll
  for (int i = 0; i < 32; i += 8)
    xTb[(size_t)(n0 + ty + i) * C_IN + (c0 + tx)] = tile[tx][ty + i];
}

// ---------------------------------------------------------------------------
// Kernel 2: repack W [o][c*7+j] -> Wt [o][j*128+c]  (k-dim = j*128 + c)
// ---------------------------------------------------------------------------
__global__ __launch_bounds__(256)
void repack_w_kernel(const float* __restrict__ W, float* __restrict__ Wt) {
  int tid = blockIdx.x * blockDim.x + threadIdx.x;
  if (tid < C_OUT * KD) {
    int o = tid / KD, rem = tid % KD;
    int j = rem >> 7, c = rem & 127;
    Wt[tid] = W[o * KD + c * KNBR + j];
  }
}

// ---------------------------------------------------------------------------
// Kernel 3: main WMMA GEMM.
//   WG = 16 waves, tile = 256(o) x 64(n), full K=896 reduction.
//   Wave w: o-range [w*16, w*16+16), 4 accumulators covering n-subtiles of 16.
//   A (16x4 f32): lane l holds Wt[o0 + l%16][k + 2*(l/16) ..+1]  (b64 load)
//   B (4x16 f32): lane l holds F[n = nt*16 + l%16][k + 2*(l/16) ..+1] from LDS
//   D (16x16 f32): vgpr r -> o = o0 + 8*(l/16) + r, n = nt*16 + l%16
//   Gather for chunk j+1 is register-double-buffered behind chunk j's compute.
// ---------------------------------------------------------------------------
__global__ __launch_bounds__(512)
void meshconv_wmma_kernel(const float* __restrict__ xT,
                          const float* __restrict__ Wt,
                          const long long* __restrict__ Gi,
                          const float* __restrict__ bias,
                          float* __restrict__ out,
                          int Nv) {
  __shared__ float F[TILE_N * FSTR];          // 64 x 132 dwords = 33.8 KB

  const int b  = blockIdx.y;
  const int n0 = blockIdx.x * TILE_N;
  const int t    = threadIdx.x;               // 0..511
  const int lane = t & 31;
  const int wave = t >> 5;                    // 0..15
  const int h    = lane >> 4;                 // half-wave selector
  const int l    = lane & 15;
  const int o0   = wave * 16;

  // gather-fill assignment: 8 threads per vertex, 16 floats each
  const int gn = t >> 3;                      // vertex 0..63
  const int gq = t & 7;                       // 0..7 within vertex

  // preload this thread's vertex neighbor indices (read-once stream)
  const long long* gptr = Gi + ((long long)b * Nv + (n0 + gn)) * KNBR;
  long long idxs[KNBR];
#pragma unroll
  for (int j = 0; j < KNBR; ++j) idxs[j] = __builtin_nontemporal_load(&gptr[j]);

  const float* xTb = xT + ((size_t)b * Nv << 7) + gq * 4;

  // prologue: gather chunk 0 into registers
  v4f reg[4];
  {
    long long idx = idxs[0];
    if (idx >= 0 && idx < Nv) {
      const float* src = xTb + ((size_t)idx << 7);
#pragma unroll
      for (int s = 0; s < 4; ++s) reg[s] = *(const v4f*)(src + s * 32);
    } else {
#pragma unroll
      for (int s = 0; s < 4; ++s) reg[s] = (v4f){0.f, 0.f, 0.f, 0.f};
    }
  }

  v8f acc0 = {}, acc1 = {}, acc2 = {}, acc3 = {};

  float* dst = &F[gn * FSTR + gq * 4];
  const float* B0 = &F[(0 * 16 + l) * FSTR + 2 * h];
  const float* B1 = &F[(1 * 16 + l) * FSTR + 2 * h];
  const float* B2 = &F[(2 * 16 + l) * FSTR + 2 * h];
  const float* B3 = &F[(3 * 16 + l) * FSTR + 2 * h];

#pragma unroll
  for (int j = 0; j < KNBR; ++j) {
    // ---- commit staged registers for chunk j to LDS ----
#pragma unroll
    for (int s = 0; s < 4; ++s) *(v4f*)(dst + s * 32) = reg[s];
    __syncthreads();

    // ---- issue chunk j+1 gather now; its loadcnt-wait lands after compute ----
    if (j + 1 < KNBR) {
      long long idx = idxs[j + 1];
      if (idx >= 0 && idx < Nv) {
        const float* src = xTb + ((size_t)idx << 7);
#pragma unroll
        for (int s = 0; s < 4; ++s) reg[s] = *(const v4f*)(src + s * 32);
      } else {
#pragma unroll
        for (int s = 0; s < 4; ++s) reg[s] = (v4f){0.f, 0.f, 0.f, 0.f};
      }
    }

    // ---- 32 k-steps of 16x16x4 fp32 WMMA, 4 n-subtiles per wave ----
    const float* Arow = Wt + (size_t)(o0 + l) * KD + j * C_IN + 2 * h;
#pragma unroll 4
    for (int c = 0; c < C_IN; c += 4) {
      v2f a  = *(const v2f*)(Arow + c);
      v2f b0 = *(const v2f*)(B0 + c);
      v2f b1 = *(const v2f*)(B1 + c);
      v2f b2 = *(const v2f*)(B2 + c);
      v2f b3 = *(const v2f*)(B3 + c);
      acc0 = __builtin_amdgcn_wmma_f32_16x16x4_f32(false, a, false, b0, (short)0, acc0, false, false);
      acc1 = __builtin_amdgcn_wmma_f32_16x16x4_f32(false, a, false, b1, (short)0, acc1, false, false);
      acc2 = __builtin_amdgcn_wmma_f32_16x16x4_f32(false, a, false, b2, (short)0, acc2, false, false);
      acc3 = __builtin_amdgcn_wmma_f32_16x16x4_f32(false, a, false, b3, (short)0, acc3, false, false);
    }
    __syncthreads();
  }

  // ---- bias + store (nontemporal: out is write-once, keep L2 for x/W) ----
  float bs[8];
#pragma unroll
  for (int r = 0; r < 8; ++r) bs[r] = bias[o0 + 8 * h + r];

#pragma unroll
  for (int r = 0; r < 8; ++r) {
    float* base = out + ((size_t)b * C_OUT + (o0 + 8 * h + r)) * (size_t)Nv + n0 + l;
    __builtin_nontemporal_store(acc0[r] + bs[r], base + 0);
    __builtin_nontemporal_store(acc1[r] + bs[r], base + 16);
    __builtin_nontemporal_store(acc2[r] + bs[r], base + 32);
    __builtin_nontemporal_store(acc3[r] + bs[r], base + 48);
  }
}

// ---------------------------------------------------------------------------
// Fallback (only if workspace is too small): naive scalar kernel.
// ---------------------------------------------------------------------------
__global__ __launch_bounds__(256)
void meshconv_naive_kernel(const float* __restrict__ x,
                           const long long* __restrict__ Gi,
                           const float* __restrict__ W,
                           const float* __restrict__ bias,
                           float* __restrict__ out,
                           int Nv) {
  long long gid = (long long)blockIdx.x * blockDim.x + threadIdx.x;
  long long total = (long long)4 * C_OUT * Nv;
  if (gid >= total) return;
  int n = (int)(gid % Nv);
  int o = (int)((gid / Nv) % C_OUT);
  int b = (int)(gid / ((long long)Nv * C_OUT));
  long long idx[KNBR];
#pragma unroll
  for (int j = 0; j < KNBR; ++j)
    idx[j] = Gi[((long long)b * Nv + n) * KNBR + j];
  float sum = bias[o];
  for (int c = 0; c < C_IN; ++c) {
    const float* xc = x + ((size_t)b * C_IN + c) * Nv;
    const float* wc = W + ((size_t)o * C_IN + c) * KNBR;
#pragma unroll
    for (int j = 0; j < KNBR; ++j)
      if (idx[j] >= 0 && idx[j] < Nv) sum += xc[idx[j]] * wc[j];
  }
  out[gid] = sum;
}

// ---------------------------------------------------------------------------
extern "C" void kernel_launch(void* const* d_in, const int* in_sizes, int n_in,
                              void* d_out, int out_size, void* d_ws, size_t ws_size,
                              hipStream_t stream) {
  const float*     x    = (const float*)d_in[0];      // [4,128,32768,1] fp32
  const long long* Gi   = (const long long*)d_in[1];  // [4,32768,7]     int64
  const float*     W    = (const float*)d_in[2];      // [256,128,1,7]   fp32
  const float*     bias = (const float*)d_in[3];      // [256]           fp32
  float*           out  = (float*)d_out;              // [4,256,32768,1] fp32

  const int B = 4, Nv = 32768;
  const size_t xT_elems = (size_t)B * Nv * C_IN;       // 16,777,216 floats
  const size_t wt_elems = (size_t)C_OUT * KD;          //    229,376 floats
  const size_t need = (xT_elems + wt_elems) * sizeof(float);

  if (ws_size >= need) {
    float* xT = (float*)d_ws;
    float* Wt = xT + xT_elems;
    transpose_x_kernel<<<dim3(Nv / 32, C_IN / 32, B), dim3(32, 8), 0, stream>>>(x, xT, Nv);
    repack_w_kernel<<<(C_OUT * KD + 255) / 256, 256, 0, stream>>>(W, Wt);
    meshconv_wmma_kernel<<<dim3(Nv / TILE_N, B), 512, 0, stream>>>(xT, Wt, Gi, bias, out, Nv);
  } else {
    long long total = (long long)B * C_OUT * Nv;
    meshconv_naive_kernel<<<(int)((total + 255) / 256), 256, 0, stream>>>(x, Gi, W, bias, out, Nv);
  }
}